// ScaledDotProductAttention_2671469658564
// MI455X (gfx1250) — compile-verified
//
#include <hip/hip_runtime.h>
#include <hip/hip_bf16.h>
#include <math.h>

// CDNA5 (gfx1250) wave32 WMMA fragments for V_WMMA_F32_16X16X4_F32:
//   A (16x4 f32)  -> 2 VGPRs/lane  -> v2f
//   B (4x16 f32)  -> 2 VGPRs/lane  -> v2f
//   C/D (16x16)   -> 8 VGPRs/lane  -> v8f
typedef __attribute__((ext_vector_type(2))) float v2f;
typedef __attribute__((ext_vector_type(4))) float v4f;
typedef __attribute__((ext_vector_type(8))) float v8f;

#define B_   2
#define H_   16
#define S_   2048
#define D_   64
#define QT   16            // query rows per tile (one WMMA M-tile)
#define NQT  (S_ / QT)     // 128 query tiles per (b,h)
#define NWAVE 4            // waves per workgroup
#define TEMP_INV 0.125f    // 1 / TEMPERATURE(=8)
#define SENT  (-1.0e30f)   // masked-score sentinel (finite, avoids inf-inf NaN)

__global__ __launch_bounds__(NWAVE * 32)
void sdpa_f32_wmma_kernel(const float* __restrict__ qg,
                          const float* __restrict__ kg,
                          const float* __restrict__ vg,
                          const int*   __restrict__ maskg,
                          float* __restrict__ outg,    // [B,H,S,D]
                          float* __restrict__ attng)   // [B,H,S,S]
{
    // Dynamic LDS: scores P[16][2048] (128 KB) + partial O[4][16][64] (16 KB)
    extern __shared__ float sP[];
    float* sO = sP + QT * S_;

    const int tid  = threadIdx.x;
    const int w    = tid >> 5;        // wave id 0..3
    const int lane = tid & 31;        // lane 0..31
    const int half = lane >> 4;       // 0: lanes 0-15, 1: lanes 16-31
    const int mrow = lane & 15;       // row/col index within half

    const int bid = blockIdx.x;
    const int qt  = bid % NQT;
    const int h   = (bid / NQT) % H_;
    const int b   = bid / (NQT * H_);
    const int q0  = qt * QT;

    const size_t bh  = ((size_t)b * H_ + h) * (size_t)S_ * D_;
    const float* qp  = qg + bh;
    const float* kp  = kg + bh;
    const float* vp  = vg + bh;
    const int*   mp  = maskg + (size_t)b * S_ * S_;

    // ---- Q tile A-fragments: 16 d-slices of 16(M) x 4(K) ----
    // A layout: lanes 0-15 hold M=lane (K = 4s+0, 4s+1); lanes 16-31 hold
    // M=lane-16 (K = 4s+2, 4s+3)  =>  contiguous v2f load per lane.
    v2f qa[16];
    #pragma unroll
    for (int s = 0; s < 16; ++s) {
        qa[s] = *(const v2f*)(qp + (size_t)(q0 + mrow) * D_ + 4 * s + 2 * half);
    }

    // ================= Pass 1: S = mask(Q K^T / T) -> LDS =================
    // Each wave takes key-chunks kc = w, w+4, ... (disjoint LDS columns).
    for (int kc = w; kc < S_ / 16; kc += NWAVE) {
        v8f c = {};
        #pragma unroll
        for (int s = 0; s < 16; ++s) {
            // B(k=d, n=key): b.x = K[key=kc*16+n][d=4s+2*half], b.y = d+1
            v2f bf = *(const v2f*)(kp + (size_t)(kc * 16 + mrow) * D_ + 4 * s + 2 * half);
            c = __builtin_amdgcn_wmma_f32_16x16x4_f32(false, qa[s], false, bf,
                                                      (short)0, c, false, false);
        }
        // C/D layout: VGPR r, lanes 0-15 -> (M=r,   N=lane),
        //                     lanes16-31 -> (M=r+8, N=lane-16)
        const int kcol = kc * 16 + mrow;
        #pragma unroll
        for (int r = 0; r < 8; ++r) {
            const int m    = r + 8 * half;
            const int qrow = q0 + m;
            float sv = c[r] * TEMP_INV;
            if (mp[(size_t)qrow * S_ + kcol] == 0) sv = SENT;
            sP[m * S_ + kcol] = sv;
        }
    }
    __syncthreads();

    // ================= Pass 2: row softmax + attn store =================
    // Wave w handles rows [4w, 4w+4); 32 lanes x float4 stride the 2048 cols
    // (ds_load_b128 / global_store_b128: 512 B per wave-op on the 536 MB
    //  attn stream that dominates the roofline).
    for (int m = w * 4; m < w * 4 + 4; ++m) {
        float* row = sP + m * S_;

        float mx = SENT;
        for (int cix = lane * 4; cix < S_; cix += 128) {
            v4f sv = *(const v4f*)(row + cix);
            mx = fmaxf(mx, fmaxf(fmaxf(sv.x, sv.y), fmaxf(sv.z, sv.w)));
        }
        #pragma unroll
        for (int off = 16; off > 0; off >>= 1)
            mx = fmaxf(mx, __shfl_xor(mx, off, 32));

        float sum = 0.0f;
        for (int cix = lane * 4; cix < S_; cix += 128) {
            v4f sv = *(const v4f*)(row + cix);
            v4f e;
            e.x = (sv.x <= SENT) ? 0.0f : __expf(sv.x - mx);
            e.y = (sv.y <= SENT) ? 0.0f : __expf(sv.y - mx);
            e.z = (sv.z <= SENT) ? 0.0f : __expf(sv.z - mx);
            e.w = (sv.w <= SENT) ? 0.0f : __expf(sv.w - mx);
            *(v4f*)(row + cix) = e;
            sum += (e.x + e.y) + (e.z + e.w);
        }
        #pragma unroll
        for (int off = 16; off > 0; off >>= 1)
            sum += __shfl_xor(sum, off, 32);

        const float inv = (sum > 0.0f) ? 1.0f / sum : 0.0f;
        float* arow = attng + (((size_t)b * H_ + h) * S_ + (size_t)(q0 + m)) * S_;
        for (int cix = lane * 4; cix < S_; cix += 128) {
            v4f p = *(const v4f*)(row + cix) * inv;
            *(v4f*)(row + cix) = p;                          // keep P for pass 3
            // attn written once, never re-read: non-temporal store keeps the
            // L2-resident K/V/mask working set from being evicted.
            __builtin_nontemporal_store(p, (v4f*)(arow + cix));
        }
    }
    __syncthreads();

    // ================= Pass 3: O = P @ V =================
    // Each wave accumulates a K-partial over ks = w, w+4, ... (4 keys/slice).
    v8f acc[4] = {};
    for (int ks = w; ks < S_ / 4; ks += NWAVE) {
        const int kbase = 4 * ks + 2 * half;
        // A from LDS (P): a.x = P[M=mrow][kbase], a.y = P[mrow][kbase+1]
        v2f a = *(const v2f*)(sP + mrow * S_ + kbase);
        const float* vrow = vp + (size_t)kbase * D_;
        #pragma unroll
        for (int t = 0; t < 4; ++t) {   // 4 N-tiles cover D=64
            v2f bf;
            bf.x = vrow[16 * t + mrow];        // V[kbase    ][16t+n]
            bf.y = vrow[D_ + 16 * t + mrow];   // V[kbase + 1][16t+n]
            acc[t] = __builtin_amdgcn_wmma_f32_16x16x4_f32(false, a, false, bf,
                                                           (short)0, acc[t], false, false);
        }
    }
    // Spill per-wave partials to LDS, reduce across waves, store output.
    #pragma unroll
    for (int t = 0; t < 4; ++t) {
        #pragma unroll
        for (int r = 0; r < 8; ++r) {
            const int m = r + 8 * half;
            sO[w * (QT * D_) + m * D_ + 16 * t + mrow] = acc[t][r];
        }
    }
    __syncthreads();

    for (int m = w * 4; m < w * 4 + 4; ++m) {
        // 32 lanes x v2f cover D=64 in one step
        const int cix = lane * 2;
        v2f sv = {0.0f, 0.0f};
        #pragma unroll
        for (int ww = 0; ww < NWAVE; ++ww) {
            v2f p = *(const v2f*)(sO + ww * (QT * D_) + m * D_ + cix);
            sv.x += p.x;
            sv.y += p.y;
        }
        *(v2f*)(outg + bh + (size_t)(q0 + m) * D_ + cix) = sv;
    }
}

extern "C" void kernel_launch(void* const* d_in, const int* in_sizes, int n_in,
                              void* d_out, int out_size, void* d_ws, size_t ws_size,
                              hipStream_t stream) {
    const float* q    = (const float*)d_in[0];
    const float* k    = (const float*)d_in[1];
    const float* v    = (const float*)d_in[2];
    const int*   mask = (const int*)d_in[3];

    float* out  = (float*)d_out;                              // [B,H,S,D] first
    float* attn = out + (size_t)B_ * H_ * S_ * D_;            // then [B,H,S,S]

    const dim3 grid(B_ * H_ * NQT);   // 4096 query tiles
    const dim3 block(NWAVE * 32);     // 4 wave32 per workgroup
    const size_t shmem = (size_t)(QT * S_ + NWAVE * QT * D_) * sizeof(float); // 144 KB

    sdpa_f32_wmma_kernel<<<grid, block, shmem, stream>>>(q, k, v, mask, out, attn);
}